// Texture_26474178413072
// MI455X (gfx1250) — compile-verified
//
#include <hip/hip_runtime.h>
#include <hip/hip_bf16.h>

// ---------------------------------------------------------------------------
// Multi-level bilinear grid sample, border padding, align_corners=False.
//   grid: (4, 512, 512, 2) f32    textures: (16, S, S) f32, S=1024/512/256/128
//   out:  (4, 16, 512, 512) f32 = sum of samples over the 4 levels
// Gather-bound kernel: textures (85MB) live in the 192MB L2; one thread per
// output pixel computes all 16 channels. x-adjacent taps are fused into one
// global_load_b64 (border handled by address clamp + lane select, no OOB).
// The streaming grid coords are staged through LDS with the gfx1250 async
// load path (ASYNCcnt) when the builtin is available.
// ---------------------------------------------------------------------------

#define TEX_F   16
#define OUT_HW  512
#define OUT_PIX (4u * 512u * 512u)   // 1,048,576 pixels
#define BLOCK   256

typedef float vf2_base __attribute__((ext_vector_type(2)));
typedef vf2_base vf2 __attribute__((aligned(4)));   // allow dword-aligned b64 loads

#if defined(__AMDGCN__)
typedef int v2i __attribute__((vector_size(8)));     // matches builtin param type
typedef __attribute__((address_space(1))) v2i* gptr_v2i;
typedef __attribute__((address_space(3))) v2i* lptr_v2i;
#endif

__device__ __forceinline__ void sample_level_accum(const float* __restrict__ tex,
                                                   const int S,
                                                   const float gx, const float gy,
                                                   float* __restrict__ acc) {
    const float Sf = (float)S;
    // unnormalize + border clamp
    float ix = ((gx + 1.0f) * Sf - 1.0f) * 0.5f;
    float iy = ((gy + 1.0f) * Sf - 1.0f) * 0.5f;
    ix = fminf(fmaxf(ix, 0.0f), Sf - 1.0f);
    iy = fminf(fmaxf(iy, 0.0f), Sf - 1.0f);

    const float x0f = floorf(ix);
    const float y0f = floorf(iy);
    const float wx = ix - x0f;
    const float wy = iy - y0f;

    const int x0  = (int)x0f;
    const int y0  = (int)y0f;
    const int y1  = min(y0 + 1, S - 1);
    const int xlo = min(x0, S - 2);        // clamp so the b64 load stays in-row
    const bool hi = (x0 > xlo);            // x0 == S-1: both taps are the .y lane

    const unsigned row0 = (unsigned)(y0 * S + xlo);
    const unsigned row1 = (unsigned)(y1 * S + xlo);
    const unsigned chan = (unsigned)(S * S);

#pragma unroll
    for (int f = 0; f < TEX_F; ++f) {
        const unsigned fo = (unsigned)f * chan;
        const vf2 p0 = *reinterpret_cast<const vf2*>(tex + fo + row0);
        const vf2 p1 = *reinterpret_cast<const vf2*>(tex + fo + row1);
        const float v00 = hi ? p0.y : p0.x;
        const float v01 = p0.y;
        const float v10 = hi ? p1.y : p1.x;
        const float v11 = p1.y;
        const float top = fmaf(wx, v01 - v00, v00);
        const float bot = fmaf(wx, v11 - v10, v10);
        acc[f] = fmaf(wy, bot - top, top) + acc[f];
    }
}

__global__ __launch_bounds__(BLOCK) void
texture_mlvl_grid_sample(const float* __restrict__ grid,
                         const float* __restrict__ t1,
                         const float* __restrict__ t2,
                         const float* __restrict__ t3,
                         const float* __restrict__ t4,
                         float* __restrict__ out) {
    const unsigned idx = blockIdx.x * BLOCK + threadIdx.x;   // exact fit: 4096*256

    float gx, gy;
#if defined(__AMDGCN__) && defined(__gfx1250__) && \
    __has_builtin(__builtin_amdgcn_global_load_async_to_lds_b64)
    // gfx1250 async path: stage this block's grid coords (8B/lane) into LDS.
    __shared__ float sgrid[BLOCK * 2];
    {
        gptr_v2i gp = (gptr_v2i)(grid + (size_t)idx * 2u);          // strips const
        lptr_v2i lp = (lptr_v2i)(&sgrid[threadIdx.x * 2u]);
        __builtin_amdgcn_global_load_async_to_lds_b64(gp, lp, 0, 0);
        asm volatile("s_wait_asynccnt 0" ::: "memory");   // ASYNCcnt==0 + reorder fence
        gx = sgrid[threadIdx.x * 2u + 0u];
        gy = sgrid[threadIdx.x * 2u + 1u];
    }
#else
    {
        const float2 g = reinterpret_cast<const float2*>(grid)[idx];
        gx = g.x;
        gy = g.y;
    }
#endif

    float acc[TEX_F];
#pragma unroll
    for (int f = 0; f < TEX_F; ++f) acc[f] = 0.0f;

    sample_level_accum(t1, 1024, gx, gy, acc);
    sample_level_accum(t2,  512, gx, gy, acc);
    sample_level_accum(t3,  256, gx, gy, acc);
    sample_level_accum(t4,  128, gx, gy, acc);

    // out layout (B, F, H, W): (b*16+f)*512*512 + h*512 + w
    const unsigned hw   = idx & (512u * 512u - 1u);
    const unsigned b    = idx >> 18;
    const unsigned base = b * (TEX_F * 512u * 512u) + hw;
#pragma unroll
    for (int f = 0; f < TEX_F; ++f) {
        out[base + (unsigned)f * (512u * 512u)] = acc[f];
    }
}

extern "C" void kernel_launch(void* const* d_in, const int* in_sizes, int n_in,
                              void* d_out, int out_size, void* d_ws, size_t ws_size,
                              hipStream_t stream) {
    (void)in_sizes; (void)n_in; (void)out_size; (void)d_ws; (void)ws_size;
    const float* grid = (const float*)d_in[0];
    const float* t1   = (const float*)d_in[1];
    const float* t2   = (const float*)d_in[2];
    const float* t3   = (const float*)d_in[3];
    const float* t4   = (const float*)d_in[4];
    float* out        = (float*)d_out;

    dim3 grd(OUT_PIX / BLOCK);   // 4096 blocks
    dim3 blk(BLOCK);
    hipLaunchKernelGGL(texture_mlvl_grid_sample, grd, blk, 0, stream,
                       grid, t1, t2, t3, t4, out);
}